// PixelShuffleTorch_17952963297514
// MI455X (gfx1250) — compile-verified
//
#include <hip/hip_runtime.h>

// PixelShuffle (torch ordering), r = 2, NHWC fp32.
// in : (8, 128, 128, 256)  out : (8, 256, 256, 64)
// out[b, 2h+dx, 2w+dy, c] = in[b, h, w, c*4 + dx*2 + dy]
//
// Pure data movement: 128 MiB in + 128 MiB out => ~11.5 us roofline @ 23.3 TB/s.
// Strategy: each thread owns 4 output channels of one input pixel.
//   - 4x global_load_b128 (16 contiguous input floats, NT hint)
//   - register regroup (the "transpose" happens inside the thread)
//   - 4x global_store_b128 into the four (dx,dy) corners, each a fully
//     written 256B-contiguous chunk per 16-thread pixel group (NT hint).
// No LDS, no WMMA: zero-FLOP permutation, so the matrix pipe cannot help;
// the win comes from b128 both ways + non-temporal TH bits (gfx1250 cache
// policy) so the 256 MiB stream does not thrash the 192 MB L2.

typedef __attribute__((ext_vector_type(4))) float v4f;

__global__ __launch_bounds__(256)
void pixel_shuffle_r2_nhwc(const float* __restrict__ in,
                           float* __restrict__ out,
                           int npix /* = B*H*W = 131072 */) {
    const int gid = blockIdx.x * blockDim.x + threadIdx.x;
    const int p = gid >> 4;   // input pixel index (b,h,w) flattened
    const int t = gid & 15;   // 4-output-channel group: c = 4t .. 4t+3
    if (p >= npix) return;

    // ---- load: 16 consecutive floats of this pixel's 256-channel vector ----
    // addr = p*256 + t*16 floats, 64 B aligned -> four b128 loads
    const v4f* __restrict__ src =
        (const v4f*)(in + (((size_t)p) << 8) + ((size_t)t << 4));
    v4f q0 = __builtin_nontemporal_load(src + 0);  // c=4t+0, offs (0,0)(0,1)(1,0)(1,1)
    v4f q1 = __builtin_nontemporal_load(src + 1);  // c=4t+1
    v4f q2 = __builtin_nontemporal_load(src + 2);  // c=4t+2
    v4f q3 = __builtin_nontemporal_load(src + 3);  // c=4t+3

    // ---- decode (b,h,w); all power-of-two -> shifts/ands only ----
    const int b = p >> 14;          // / (128*128)
    const int h = (p >> 7) & 127;   // (/128) % 128
    const int w = p & 127;          // % 128

    // out flat index: ((b*256 + 2h+dx)*256 + 2w+dy)*64 + c
    //  = b<<22 | h<<15 | w<<7 | 4t   (+ dx*16384 + dy*64)
    float* __restrict__ o = out + (((size_t)b << 22) + ((size_t)h << 15) +
                                   ((size_t)w << 7) + ((size_t)t << 2));

    // ---- regroup by output corner (dx,dy) ----
    v4f o00 = {q0.x, q1.x, q2.x, q3.x};  // (dx=0,dy=0)
    v4f o01 = {q0.y, q1.y, q2.y, q3.y};  // (dx=0,dy=1)
    v4f o10 = {q0.z, q1.z, q2.z, q3.z};  // (dx=1,dy=0)
    v4f o11 = {q0.w, q1.w, q2.w, q3.w};  // (dx=1,dy=1)

    // ---- store: four b128 stores; per pixel each corner is a contiguous,
    // fully-written 256 B chunk -> whole cache lines, no RMW ----
    __builtin_nontemporal_store(o00, (v4f*)(o));
    __builtin_nontemporal_store(o01, (v4f*)(o + 64));            // dy=1
    __builtin_nontemporal_store(o10, (v4f*)(o + 16384));         // dx=1
    __builtin_nontemporal_store(o11, (v4f*)(o + 16384 + 64));    // dx=1,dy=1
}

extern "C" void kernel_launch(void* const* d_in, const int* in_sizes, int n_in,
                              void* d_out, int out_size, void* d_ws, size_t ws_size,
                              hipStream_t stream) {
    (void)n_in; (void)out_size; (void)d_ws; (void)ws_size;
    const float* x = (const float*)d_in[0];
    float* out = (float*)d_out;

    const int npix  = in_sizes[0] / 256;   // 8*128*128 = 131072 pixels
    const int total = npix * 16;           // 16 threads per pixel
    const int block = 256;                 // 8 wave32s per workgroup
    const int grid  = (total + block - 1) / block;  // 8192 blocks

    pixel_shuffle_r2_nhwc<<<grid, block, 0, stream>>>(x, out, npix);
}